// BraskModel_31241592111295
// MI455X (gfx1250) — compile-verified
//
#include <hip/hip_runtime.h>
#include <math.h>

typedef __attribute__((ext_vector_type(2))) float v2f;
typedef __attribute__((ext_vector_type(8))) float v8f;

constexpr int Bc = 2, Lc = 384, Hc = 768, Rc = 16, RDc = 256, Ac = 256, Kc = 8;

#if defined(__has_builtin)
#if __has_builtin(__builtin_amdgcn_wmma_f32_16x16x4_f32)
#define HAVE_WMMA_F32_16X16X4 1
#endif
#endif

__device__ __forceinline__ float sigmoidf_(float x) { return 1.0f / (1.0f + expf(-x)); }

// ---------------------------------------------------------------------------
// 1) Entity heads: 4 GEMV + sigmoid over rows of X (B*L rows, H each).
//    One wave per row computes all four dots.
// ---------------------------------------------------------------------------
__global__ __launch_bounds__(256)
void ent_heads_kernel(const float* __restrict__ X,
                      const float* __restrict__ w0, const float* __restrict__ b0,
                      const float* __restrict__ w1, const float* __restrict__ b1,
                      const float* __restrict__ w2, const float* __restrict__ b2,
                      const float* __restrict__ w3, const float* __restrict__ b3,
                      float* __restrict__ o0, float* __restrict__ o1,
                      float* __restrict__ o2, float* __restrict__ o3) {
  int wave = (blockIdx.x * blockDim.x + threadIdx.x) >> 5;
  int lane = threadIdx.x & 31;
  if (wave >= Bc * Lc) return;
  const float* row = X + (size_t)wave * Hc;
  float a0 = 0.f, a1 = 0.f, a2 = 0.f, a3 = 0.f;
  for (int h = lane; h < Hc; h += 32) {
    float x = row[h];
    a0 += x * w0[h]; a1 += x * w1[h]; a2 += x * w2[h]; a3 += x * w3[h];
  }
  for (int off = 16; off; off >>= 1) {
    a0 += __shfl_xor(a0, off); a1 += __shfl_xor(a1, off);
    a2 += __shfl_xor(a2, off); a3 += __shfl_xor(a3, off);
  }
  if (lane == 0) {
    o0[wave] = sigmoidf_(a0 + b0[0]);
    o1[wave] = sigmoidf_(a1 + b1[0]);
    o2[wave] = sigmoidf_(a2 + b2[0]);
    o3[wave] = sigmoidf_(a3 + b3[0]);
  }
}

// ---------------------------------------------------------------------------
// 2) fp32 WMMA GEMM: Y(M,N) = Xm(M,K) @ W(K,N) + bias(N).
//    One wave per 16x16 output tile, K stepped by 4 via v_wmma_f32_16x16x4_f32.
//    Fragment layouts per CDNA5 ISA 7.12.2:
//      A 16x4:  lanes 0-15 -> M=lane, K={0,1}; lanes 16-31 -> M=lane-16, K={2,3}
//      B 4x16:  lanes 0-15 -> N=lane, K={0,1}; lanes 16-31 -> N=lane-16, K={2,3}
//      C/D:     VGPR i -> M = i + 8*(lane>=16), N = lane&15
// ---------------------------------------------------------------------------
__global__ __launch_bounds__(32)
void wmma_gemm_kernel(const float* __restrict__ Xm, const float* __restrict__ W,
                      const float* __restrict__ bias, float* __restrict__ Y,
                      int M, int K, int N) {
  int tileM = blockIdx.x, tileN = blockIdx.y;
  int lane = threadIdx.x;
  int m16 = lane & 15;
  int kb = (lane >> 4) * 2;
  const float* Arow = Xm + (size_t)(tileM * 16 + m16) * K;
  const float* Bcol = W + (tileN * 16 + m16);
  v8f acc = {};
#if HAVE_WMMA_F32_16X16X4
#pragma unroll 4
  for (int k = 0; k < K; k += 4) {
    v2f a, b;
    a.x = Arow[k + kb];
    a.y = Arow[k + kb + 1];
    b.x = Bcol[(size_t)(k + kb) * N];
    b.y = Bcol[(size_t)(k + kb + 1) * N];
    acc = __builtin_amdgcn_wmma_f32_16x16x4_f32(false, a, false, b, (short)0, acc,
                                                false, false);
  }
#else
  // Fallback (should not be taken on gfx1250): scalar accumulate of own outputs.
  for (int k = 0; k < K; ++k) {
    float bv = W[(size_t)k * N + tileN * 16 + m16];
    for (int i = 0; i < 8; ++i) {
      int rr = tileM * 16 + i + 8 * (lane >> 4);
      acc[i] += Xm[(size_t)rr * K + k] * bv;
    }
  }
#endif
  int col = tileN * 16 + m16;
  float bv = bias ? bias[col] : 0.f;
  int rbase = tileM * 16 + 8 * (lane >> 4);
#pragma unroll
  for (int i = 0; i < 8; ++i)
    Y[(size_t)(rbase + i) * N + col] = acc[i] + bv;
}

// ---------------------------------------------------------------------------
// 3) Tiny naive GEMM+bias: out(M,N) = A(M,K) @ W(K,N) + b(N). 1 thread / output.
// ---------------------------------------------------------------------------
__global__ __launch_bounds__(256)
void gemm_naive_kernel(const float* __restrict__ Am, const float* __restrict__ W,
                       const float* __restrict__ bias, float* __restrict__ out,
                       int M, int K, int N) {
  int idx = blockIdx.x * blockDim.x + threadIdx.x;
  if (idx >= M * N) return;
  int i = idx / N, j = idx % N;
  float acc = bias[j];
  const float* arow = Am + (size_t)i * K;
  for (int k = 0; k < K; ++k) acc += arow[k] * W[(size_t)k * N + j];
  out[idx] = acc;
}

// ---------------------------------------------------------------------------
// 4) Relation attention: one block per (b,r).
//    e[l] = sum_a tanh(wx[b,l,a]+wr[r,a]+wg[b,a]) * v[a] + vb
//    a    = softmax_l(e);  c[b,r,h] = sum_l a[l] * X[b,l,h]
// ---------------------------------------------------------------------------
__global__ __launch_bounds__(256)
void rel_att_kernel(const float* __restrict__ wx, const float* __restrict__ wr,
                    const float* __restrict__ wg, const float* __restrict__ vvec,
                    const float* __restrict__ vb, const float* __restrict__ X,
                    float* __restrict__ cout) {
  int b = blockIdx.x >> 4;  // / Rc
  int r = blockIdx.x & 15;
  int t = threadIdx.x;
  int wave = t >> 5, lane = t & 31;
  __shared__ float sh_e[Lc];
  __shared__ float red[256];

  for (int l = wave; l < Lc; l += 8) {
    const float* wxrow = wx + ((size_t)b * Lc + l) * Ac;
    float acc = 0.f;
    for (int a = lane; a < Ac; a += 32) {
      float z = tanhf(wxrow[a] + wr[r * Ac + a] + wg[b * Ac + a]);
      acc += z * vvec[a];
    }
    for (int off = 16; off; off >>= 1) acc += __shfl_xor(acc, off);
    if (lane == 0) sh_e[l] = acc + vb[0];
  }
  __syncthreads();

  float mx = -3.4e38f;
  for (int l = t; l < Lc; l += 256) mx = fmaxf(mx, sh_e[l]);
  red[t] = mx; __syncthreads();
  for (int s = 128; s; s >>= 1) { if (t < s) red[t] = fmaxf(red[t], red[t + s]); __syncthreads(); }
  mx = red[0]; __syncthreads();

  float sum = 0.f;
  for (int l = t; l < Lc; l += 256) { float ex = expf(sh_e[l] - mx); sh_e[l] = ex; sum += ex; }
  red[t] = sum; __syncthreads();
  for (int s = 128; s; s >>= 1) { if (t < s) red[t] += red[t + s]; __syncthreads(); }
  float inv = 1.f / red[0];

  for (int h = t; h < Hc; h += 256) {
    float acc = 0.f;
    for (int l = 0; l < Lc; ++l) acc += sh_e[l] * X[((size_t)b * Lc + l) * Hc + h];
    cout[((size_t)b * Rc + r) * Hc + h] = acc * inv;
  }
}

// ---------------------------------------------------------------------------
// 5) Greedy span extraction (sequential scan per batch). Ranks are unique so
//    each of the first 8 valid spans fills exactly one slot (m in {0,1}).
// ---------------------------------------------------------------------------
__global__ __launch_bounds__(256)
void extract_sk_kernel(const float* __restrict__ X,
                       const float* __restrict__ sprob, const float* __restrict__ eprob,
                       float* __restrict__ sk, float* __restrict__ m) {
  int b = blockIdx.x;
  int t = threadIdx.x;
  __shared__ int spanS[Kc], spanE[Kc], nvalid;
  __shared__ unsigned char consumed[Lc + 16];
  for (int i = t; i < Lc + 16; i += 256) consumed[i] = 0;
  __syncthreads();
  if (t == 0) {
    int cnt = 0;
    for (int s = 0; s < Lc; ++s) {
      if (!(sprob[b * Lc + s] >= 0.5f)) continue;
      int e = -1;
      for (int d = 0; d < 10; ++d) {
        int ee = s + d;
        if (ee < Lc && eprob[b * Lc + ee] >= 0.5f && !consumed[ee]) { e = ee; break; }
      }
      if (e >= 0) {
        consumed[e] = 1;
        if (cnt < Kc) { spanS[cnt] = s; spanE[cnt] = e; }
        cnt++;
      }
    }
    nvalid = cnt < Kc ? cnt : Kc;
  }
  __syncthreads();
  int nv = nvalid;
  for (int k = 0; k < Kc; ++k) {
    bool valid = k < nv;
    int s = valid ? spanS[k] : 0;
    int e = valid ? spanE[k] : 0;
    for (int h = t; h < Hc; h += 256) {
      float v = 0.f;
      if (valid)
        v = 0.5f * (X[((size_t)b * Lc + s) * Hc + h] + X[((size_t)b * Lc + e) * Hc + h]);
      sk[((size_t)b * Kc + k) * Hc + h] = v;
    }
  }
  if (t < Kc) m[b * Kc + t] = (t < nv) ? 1.f : 0.f;
}

// ---------------------------------------------------------------------------
// 6) Projection vectors per (direction, head):
//    qvec = Wx @ w + w   (folds u_x + u_X),   pvec = Ws @ w
// ---------------------------------------------------------------------------
__global__ __launch_bounds__(256)
void proj_vec_kernel(const float* __restrict__ Wx, const float* __restrict__ Ws,
                     const float* __restrict__ w, float* __restrict__ qvec,
                     float* __restrict__ pvec) {
  int wave = (blockIdx.x * blockDim.x + threadIdx.x) >> 5;
  int lane = threadIdx.x & 31;
  if (wave >= Hc) return;
  const float* rx = Wx + (size_t)wave * Hc;
  const float* rs = Ws + (size_t)wave * Hc;
  float aq = 0.f, ap = 0.f;
  for (int o = lane; o < Hc; o += 32) {
    float wv = w[o];
    aq += rx[o] * wv; ap += rs[o] * wv;
  }
  for (int off = 16; off; off >>= 1) { aq += __shfl_xor(aq, off); ap += __shfl_xor(ap, off); }
  if (lane == 0) { qvec[wave] = aq + w[wave]; pvec[wave] = ap; }
}

// consts[0] = bx.w + sb ; consts[1] = bs.w
__global__ __launch_bounds__(256)
void proj_const_kernel(const float* __restrict__ bx, const float* __restrict__ bs,
                       const float* __restrict__ w, const float* __restrict__ sb,
                       float* __restrict__ consts) {
  __shared__ float red[256];
  int t = threadIdx.x;
  float a = 0.f, c2 = 0.f;
  for (int h = t; h < Hc; h += 256) { float wv = w[h]; a += bx[h] * wv; c2 += bs[h] * wv; }
  red[t] = a; __syncthreads();
  for (int s = 128; s; s >>= 1) { if (t < s) red[t] += red[t + s]; __syncthreads(); }
  float r0 = red[0]; __syncthreads();
  red[t] = c2; __syncthreads();
  for (int s = 128; s; s >>= 1) { if (t < s) red[t] += red[t + s]; __syncthreads(); }
  if (t == 0) { consts[0] = r0 + sb[0]; consts[1] = red[0]; }
}

// ---------------------------------------------------------------------------
// 7) Generic row-dot: out[i] = mask[i] * (Rows[i].v + cadd)
// ---------------------------------------------------------------------------
__global__ __launch_bounds__(256)
void rows_dot_kernel(const float* __restrict__ Rows, const float* __restrict__ vv,
                     const float* __restrict__ caddp, int caddIdx,
                     const float* __restrict__ mask, float* __restrict__ out,
                     int nrows, int K) {
  int wave = (blockIdx.x * blockDim.x + threadIdx.x) >> 5;
  int lane = threadIdx.x & 31;
  if (wave >= nrows) return;
  const float* r = Rows + (size_t)wave * K;
  float a = 0.f;
  for (int k = lane; k < K; k += 32) a += r[k] * vv[k];
  for (int off = 16; off; off >>= 1) a += __shfl_xor(a, off);
  if (lane == 0) {
    float res = a + (caddp ? caddp[caddIdx] : 0.f);
    if (mask) res *= mask[wave];
    out[wave] = res;
  }
}

// ---------------------------------------------------------------------------
// 8) Final fused output: out[b,r,k,l] = sigmoid(tl[b,l] + uk[b,k] + uc[b,r])
// ---------------------------------------------------------------------------
__global__ __launch_bounds__(256)
void final_out_kernel(const float* __restrict__ tl, const float* __restrict__ uk,
                      const float* __restrict__ uc, float* __restrict__ out) {
  int idx = blockIdx.x * blockDim.x + threadIdx.x;
  if (idx >= Bc * Rc * Kc * Lc) return;
  int l = idx % Lc;
  int tmp = idx / Lc;
  int k = tmp % Kc; tmp /= Kc;
  int r = tmp % Rc;
  int b = tmp / Rc;
  float x = tl[b * Lc + l] + uk[b * Kc + k] + uc[b * Rc + r];
  out[idx] = sigmoidf_(x);
}

// ---------------------------------------------------------------------------
extern "C" void kernel_launch(void* const* d_in, const int* in_sizes, int n_in,
                              void* d_out_, int out_size, void* d_ws, size_t ws_size,
                              hipStream_t stream) {
  const float* X   = (const float*)d_in[0];   // (B,L,H)
  const float* Xm  = (const float*)d_in[1];   // (B,H)
  const float* sem = (const float*)d_in[3];   // (R,H)
  const float* tre = (const float*)d_in[4];   // (R,RD)
  const float* P[48];
  for (int i = 0; i < n_in && i < 48; ++i) P[i] = (const float*)d_in[i];
  float* out = (float*)d_out_;
  float* ws = (float*)d_ws;

  // Output offsets (flat, return order)
  const size_t OFF_FHS = 0;
  const size_t OFF_FHE = OFF_FHS + (size_t)Bc * Lc;
  const size_t OFF_FTS = OFF_FHE + (size_t)Bc * Lc;
  const size_t OFF_FTE = OFF_FTS + (size_t)Bc * Rc * Kc * Lc;
  const size_t OFF_BTS = OFF_FTE + (size_t)Bc * Rc * Kc * Lc;
  const size_t OFF_BTE = OFF_BTS + (size_t)Bc * Lc;
  const size_t OFF_BHS = OFF_BTE + (size_t)Bc * Lc;
  const size_t OFF_BHE = OFF_BHS + (size_t)Bc * Rc * Kc * Lc;

  // Workspace carve-up (floats): ~1.94 MB total
  float* wx_s = ws;                       // (B,L,A)
  float* wx_t = wx_s + (size_t)Bc * Lc * Ac;
  float* wr_s = wx_t + (size_t)Bc * Lc * Ac;   // (R,A)
  float* wr_t = wr_s + (size_t)Rc * Ac;
  float* wg_s = wr_t + (size_t)Rc * Ac;        // (B,A)
  float* wg_t = wg_s + (size_t)Bc * Ac;
  float* c_f  = wg_t + (size_t)Bc * Ac;        // (B,R,H)
  float* c_b  = c_f + (size_t)Bc * Rc * Hc;
  float* sk_f = c_b + (size_t)Bc * Rc * Hc;    // (B,K,H)
  float* sk_b = sk_f + (size_t)Bc * Kc * Hc;
  float* m_f  = sk_b + (size_t)Bc * Kc * Hc;   // (B,K)
  float* m_b  = m_f + (size_t)Bc * Kc;
  float* qv   = m_b + (size_t)Bc * Kc;         // 4 x H
  float* pv   = qv + 4 * (size_t)Hc;           // 4 x H
  float* cns  = pv + 4 * (size_t)Hc;           // 4 x 2
  float* tl   = cns + 8;                       // 4 x (B*L)
  float* uk   = tl + 4 * (size_t)Bc * Lc;      // 4 x (B*K)
  float* uc   = uk + 4 * (size_t)Bc * Kc;      // 4 x (B*R)

  // 1) entity heads fhp_s/e (P5..8), btp_s/e (P9..12) -> fhs,fhe,bts,bte
  ent_heads_kernel<<<96, 256, 0, stream>>>(X, P[5], P[6], P[7], P[8], P[9], P[10],
                                           P[11], P[12], out + OFF_FHS, out + OFF_FHE,
                                           out + OFF_BTS, out + OFF_BTE);

  // 2) wx = X @ Wx + bx for sra (P25,P26) and tra (P33,P34) — WMMA f32
  dim3 gw(Bc * Lc / 16, Ac / 16);  // (48,16)
  wmma_gemm_kernel<<<gw, 32, 0, stream>>>(X, P[25], P[26], wx_s, Bc * Lc, Hc, Ac);
  wmma_gemm_kernel<<<gw, 32, 0, stream>>>(X, P[33], P[34], wx_t, Bc * Lc, Hc, Ac);

  // 3) wr = rel @ Wr + br ; wg = Xm @ Wg + bg
  gemm_naive_kernel<<<(Rc * Ac + 255) / 256, 256, 0, stream>>>(sem, P[21], P[22], wr_s, Rc, Hc, Ac);
  gemm_naive_kernel<<<(Rc * Ac + 255) / 256, 256, 0, stream>>>(tre, P[29], P[30], wr_t, Rc, RDc, Ac);
  gemm_naive_kernel<<<(Bc * Ac + 255) / 256, 256, 0, stream>>>(Xm, P[23], P[24], wg_s, Bc, Hc, Ac);
  gemm_naive_kernel<<<(Bc * Ac + 255) / 256, 256, 0, stream>>>(Xm, P[31], P[32], wg_t, Bc, Hc, Ac);

  // 4) attention contexts
  rel_att_kernel<<<Bc * Rc, 256, 0, stream>>>(wx_s, wr_s, wg_s, P[27], P[28], X, c_f);
  rel_att_kernel<<<Bc * Rc, 256, 0, stream>>>(wx_t, wr_t, wg_t, P[35], P[36], X, c_b);

  // 5) span extraction (f uses fhp probs, b uses btp probs)
  extract_sk_kernel<<<Bc, 256, 0, stream>>>(X, out + OFF_FHS, out + OFF_FHE, sk_f, m_f);
  extract_sk_kernel<<<Bc, 256, 0, stream>>>(X, out + OFF_BTS, out + OFF_BTE, sk_b, m_b);

  // 6) per-combo projection vectors + scalar constants
  //    combos: 0=(fef,ftp_s) 1=(fef,ftp_e) 2=(feb,bhp_s) 3=(feb,bhp_e)
  const float* WxD[4] = {P[39], P[39], P[43], P[43]};
  const float* WsD[4] = {P[37], P[37], P[41], P[41]};
  const float* BxD[4] = {P[40], P[40], P[44], P[44]};
  const float* BsD[4] = {P[38], P[38], P[42], P[42]};
  const float* Wh[4]  = {P[13], P[15], P[17], P[19]};
  const float* Bh[4]  = {P[14], P[16], P[18], P[20]};
  float* Cdir[4]  = {c_f, c_f, c_b, c_b};
  float* Mdir[4]  = {m_f, m_f, m_b, m_b};
  float* Skdir[4] = {sk_f, sk_f, sk_b, sk_b};
  float* OutD[4]  = {out + OFF_FTS, out + OFF_FTE, out + OFF_BHS, out + OFF_BHE};

  for (int c = 0; c < 4; ++c) {
    proj_vec_kernel<<<96, 256, 0, stream>>>(WxD[c], WsD[c], Wh[c], qv + (size_t)c * Hc,
                                            pv + (size_t)c * Hc);
    proj_const_kernel<<<1, 256, 0, stream>>>(BxD[c], BsD[c], Wh[c], Bh[c], cns + 2 * c);
  }

  // 7) scalar tables tl (B*L), uk (B*K, masked), uc (B*R)
  for (int c = 0; c < 4; ++c) {
    rows_dot_kernel<<<96, 256, 0, stream>>>(X, qv + (size_t)c * Hc, cns + 2 * c, 0,
                                            (const float*)nullptr,
                                            tl + (size_t)c * Bc * Lc, Bc * Lc, Hc);
    rows_dot_kernel<<<2, 256, 0, stream>>>(Skdir[c], pv + (size_t)c * Hc, cns + 2 * c, 1,
                                           Mdir[c], uk + (size_t)c * Bc * Kc, Bc * Kc, Hc);
    rows_dot_kernel<<<4, 256, 0, stream>>>(Cdir[c], Wh[c], (const float*)nullptr, 0,
                                           (const float*)nullptr,
                                           uc + (size_t)c * Bc * Rc, Bc * Rc, Hc);
  }

  // 8) fused outputs
  const int TOT = Bc * Rc * Kc * Lc;
  for (int c = 0; c < 4; ++c) {
    final_out_kernel<<<(TOT + 255) / 256, 256, 0, stream>>>(
        tl + (size_t)c * Bc * Lc, uk + (size_t)c * Bc * Kc, uc + (size_t)c * Bc * Rc,
        OutD[c]);
  }
}